// HashTableEncoder2D_57904749085051
// MI455X (gfx1250) — compile-verified
//
#include <hip/hip_runtime.h>
#include <hip/hip_bf16.h>
#include <stdint.h>
#include <math.h>

#define LEVELS 16
#define FEAT_DIM 2

// Ns[l] = Python round(16 * 1.5^l)  (banker's rounding at l=5: 121.5 -> 122)
__constant__ float kNs[LEVELS] = {
    16.f,   24.f,   36.f,   54.f,   81.f,   122.f,  182.f,  273.f,
    410.f,  615.f,  923.f,  1384.f, 2076.f, 3114.f, 4671.f, 7006.f
};

typedef __attribute__((ext_vector_type(4))) float v4f;

// grid = (ceil(H*W4/256), LEVELS, B); block = 256.
// blockIdx.y = level, blockIdx.z = batch  -> x0/y0/seed/scale are wave-uniform
// (scalar loads + SALU). Threads walk x-groups of 4 then y: coalesced B128 NT
// stores per wave, (y,xg) decoded with shift/mask (pow2 dims).
__global__ __launch_bounds__(256)
void HashTableEncoder2D_57904749085051_kernel(
    const float* __restrict__ x0,
    const float* __restrict__ y0,
    const float2* __restrict__ tab,
    const uint32_t* __restrict__ seeds,
    const int* __restrict__ d_tile,   // complete_tile_size (device scalar)
    float* __restrict__ out,
    int H, int W,
    int w4shift, int w4mask,          // W4 = W/4 = 1<<w4shift
    uint32_t table_size, uint32_t table_mask, int is_pow2,
    int plane_threads)                // H * W4
{
    const uint32_t P1 = 2654435761u;
    const uint32_t P2 = 805459861u;

    const int l = blockIdx.y;
    const int b = blockIdx.z;

    // ---- Warm GL2 with the hash table (global_prefetch_b8). One 256B slice
    // per block; 16384 blocks cover the full 4MB table. ----
    if (threadIdx.x == 0) {
        const char* tb = (const char*)tab;
        size_t tbytes  = (size_t)table_size * sizeof(float2);
        size_t flatblk = ((size_t)b * gridDim.y + (size_t)l) * gridDim.x + blockIdx.x;
        size_t off     = flatblk * 256u;
        if (off < tbytes) {
            __builtin_prefetch(tb + off, 0, 1);
            if (off + 128u < tbytes) __builtin_prefetch(tb + off + 128u, 0, 1);
        }
    }

    int idx = blockIdx.x * blockDim.x + threadIdx.x;   // over H * W4
    if (idx >= plane_threads) return;

    int xg = idx & w4mask;
    int y  = idx >> w4shift;

    // Wave-uniform setup (scalar loads / SALU)
    float    tile  = (float)d_tile[0];
    float    scale = kNs[l] / tile;
    uint32_t seed  = seeds[l];
    float    bx0   = x0[b];
    float    by0   = y0[b];

    // y-dependent hash partials (shared by the 4 pixels)
    float fyn  = ((float)y + by0) * scale;
    float iy0f = floorf(fyn);
    int   iy0  = (int)iy0f;
    float fy   = fyn - iy0f;
    uint32_t hy0 = ((uint32_t)iy0) * P2 ^ seed;
    uint32_t hy1 = ((uint32_t)iy0 + 1u) * P2 ^ seed;

    float xb = (float)(xg << 2) + bx0;

    float o0[4], o1[4];
#pragma unroll
    for (int j = 0; j < 4; ++j) {
        float fxn  = (xb + (float)j) * scale;
        float ix0f = floorf(fxn);
        int   ix0  = (int)ix0f;
        float fx   = fxn - ix0f;

        uint32_t hx0 = ((uint32_t)ix0) * P1;
        uint32_t hx1 = hx0 + P1;        // (ix0+1)*P1 mod 2^32

        uint32_t i00, i10, i01, i11;
        if (is_pow2) {
            i00 = (hx0 ^ hy0) & table_mask;
            i10 = (hx1 ^ hy0) & table_mask;
            i01 = (hx0 ^ hy1) & table_mask;
            i11 = (hx1 ^ hy1) & table_mask;
        } else {
            i00 = (hx0 ^ hy0) % table_size;
            i10 = (hx1 ^ hy0) % table_size;
            i01 = (hx0 ^ hy1) % table_size;
            i11 = (hx1 ^ hy1) % table_size;
        }

        // 4x global_load_b64 gathers (L2-resident table)
        float2 f00 = tab[i00];
        float2 f10 = tab[i10];
        float2 f01 = tab[i01];
        float2 f11 = tab[i11];

        float w00 = (1.f - fx) * (1.f - fy);
        float w10 = fx * (1.f - fy);
        float w01 = (1.f - fx) * fy;
        float w11 = fx * fy;

        o0[j] = w00 * f00.x + w10 * f10.x + w01 * f01.x + w11 * f11.x;
        o1[j] = w00 * f00.y + w10 * f10.y + w01 * f01.y + w11 * f11.y;
    }

    // out[b][2l+f][y][x] : two B128 non-temporal streaming stores
    size_t plane = (size_t)H * (size_t)W;
    size_t base  = (((size_t)b * (LEVELS * FEAT_DIM) + (size_t)(2 * l)) * (size_t)H
                    + (size_t)y) * (size_t)W + (size_t)(xg << 2);

    v4f v0 = { o0[0], o0[1], o0[2], o0[3] };
    v4f v1 = { o1[0], o1[1], o1[2], o1[3] };
    __builtin_nontemporal_store(v0, (v4f*)(out + base));
    __builtin_nontemporal_store(v1, (v4f*)(out + base + plane));
}

// Generic fallback (runtime div/mod decode) for non-pow2 crop sizes.
__global__ __launch_bounds__(256)
void HashTableEncoder2D_57904749085051_kernel_generic(
    const float* __restrict__ x0,
    const float* __restrict__ y0,
    const float2* __restrict__ tab,
    const uint32_t* __restrict__ seeds,
    const int* __restrict__ d_tile,
    float* __restrict__ out,
    int H, int W,
    uint32_t table_size, uint32_t table_mask, int is_pow2,
    int plane_threads)
{
    const uint32_t P1 = 2654435761u;
    const uint32_t P2 = 805459861u;
    const int l = blockIdx.y;
    const int b = blockIdx.z;

    int idx = blockIdx.x * blockDim.x + threadIdx.x;
    if (idx >= plane_threads) return;   // plane_threads = H * W here (1 px/thread)
    int x = idx % W;
    int y = idx / W;

    float    tile  = (float)d_tile[0];
    float    scale = kNs[l] / tile;
    uint32_t seed  = seeds[l];

    float fyn  = ((float)y + y0[b]) * scale;
    float iy0f = floorf(fyn);
    float fy   = fyn - iy0f;
    uint32_t hy0 = ((uint32_t)(int)iy0f) * P2 ^ seed;
    uint32_t hy1 = ((uint32_t)(int)iy0f + 1u) * P2 ^ seed;

    float fxn  = ((float)x + x0[b]) * scale;
    float ix0f = floorf(fxn);
    float fx   = fxn - ix0f;
    uint32_t hx0 = ((uint32_t)(int)ix0f) * P1;
    uint32_t hx1 = hx0 + P1;

    uint32_t i00, i10, i01, i11;
    if (is_pow2) {
        i00 = (hx0 ^ hy0) & table_mask; i10 = (hx1 ^ hy0) & table_mask;
        i01 = (hx0 ^ hy1) & table_mask; i11 = (hx1 ^ hy1) & table_mask;
    } else {
        i00 = (hx0 ^ hy0) % table_size; i10 = (hx1 ^ hy0) % table_size;
        i01 = (hx0 ^ hy1) % table_size; i11 = (hx1 ^ hy1) % table_size;
    }

    float2 f00 = tab[i00], f10 = tab[i10], f01 = tab[i01], f11 = tab[i11];
    float w00 = (1.f - fx) * (1.f - fy), w10 = fx * (1.f - fy);
    float w01 = (1.f - fx) * fy,         w11 = fx * fy;

    size_t plane = (size_t)H * (size_t)W;
    size_t base  = (((size_t)b * (LEVELS * FEAT_DIM) + (size_t)(2 * l)) * (size_t)H
                    + (size_t)y) * (size_t)W + (size_t)x;
    __builtin_nontemporal_store(w00*f00.x + w10*f10.x + w01*f01.x + w11*f11.x,
                                out + base);
    __builtin_nontemporal_store(w00*f00.y + w10*f10.y + w01*f01.y + w11*f11.y,
                                out + base + plane);
}

extern "C" void kernel_launch(void* const* d_in, const int* in_sizes, int n_in,
                              void* d_out, int out_size, void* d_ws, size_t ws_size,
                              hipStream_t stream) {
    (void)n_in; (void)d_ws; (void)ws_size;

    const float*    x0     = (const float*)d_in[0];
    const float*    y0     = (const float*)d_in[1];
    const float2*   tab    = (const float2*)d_in[2];
    const uint32_t* seeds  = (const uint32_t*)d_in[3];
    // d_in[4] = memorized_crop_size (device scalar) — H derived from out_size
    // instead, since grid dims must be known host-side under graph capture.
    const int*      d_tile = (const int*)d_in[5];
    float*          out    = (float*)d_out;

    int B = in_sizes[0];
    uint32_t table_size = (uint32_t)(in_sizes[2] / FEAT_DIM);
    uint32_t table_mask = table_size - 1u;
    int is_pow2 = (table_size != 0u) && ((table_size & table_mask) == 0u);

    long long hw = (long long)out_size / ((long long)B * LEVELS * FEAT_DIM);
    int H = (int)(sqrt((double)hw) + 0.5);
    int W = H;

    bool fast = (W % 4 == 0);
    int  W4   = W >> 2;
    bool w4pow2 = fast && W4 > 0 && ((W4 & (W4 - 1)) == 0);

    if (w4pow2) {
        int w4shift = 0;
        while ((1 << w4shift) < W4) ++w4shift;
        int plane_threads = H * W4;
        dim3 grid((plane_threads + 255) / 256, LEVELS, B);
        HashTableEncoder2D_57904749085051_kernel<<<grid, 256, 0, stream>>>(
            x0, y0, tab, seeds, d_tile, out,
            H, W, w4shift, W4 - 1,
            table_size, table_mask, is_pow2, plane_threads);
    } else {
        int plane_threads = H * W;
        dim3 grid((plane_threads + 255) / 256, LEVELS, B);
        HashTableEncoder2D_57904749085051_kernel_generic<<<grid, 256, 0, stream>>>(
            x0, y0, tab, seeds, d_tile, out,
            H, W, table_size, table_mask, is_pow2, plane_threads);
    }
}